// SMPLLayer_37649683317220
// MI455X (gfx1250) — compile-verified
//
#include <hip/hip_runtime.h>
#include <cmath>

#define B_ 1024
#define V_ 6890
#define J_ 24
#define NB_ 10
#define KP 208               // pose-feature K padded (207 + 1 zero)
#define VC (V_ * 3)          // posedirs row stride
#define RTS 384              // RTt floats per batch: 16 entries x 24 joints

typedef float v2f __attribute__((ext_vector_type(2)));
typedef float v8f __attribute__((ext_vector_type(8)));

// Async global->LDS copy of 16 bytes (CDNA5, tracked by ASYNCcnt).
// lds_off = wave-relative LDS byte offset (flat-address low 32 bits).
__device__ __forceinline__ void async_b128(unsigned lds_off, const void* g) {
  asm volatile("global_load_async_to_lds_b128 %0, %1, off"
               :: "v"(lds_off), "v"((unsigned long long)(uintptr_t)g)
               : "memory");
}
__device__ __forceinline__ void wait_async0() {
  asm volatile("s_wait_asynccnt 0" ::: "memory");
}
__device__ __forceinline__ unsigned lds_off32(const void* p) {
  return (unsigned)(uintptr_t)p;   // flat LDS addr[31:0] == LDS offset
}

// ---------------------------------------------------------------------------
// Kernel 1: fold J_regressor into the shape basis.
//   JS[j][c][l] = sum_v Jreg[j,v] * shapedirs[v,c,l]   (24*3*10)
//   JT[j][c]    = sum_v Jreg[j,v] * v_template[v,c]    (24*3)
// ---------------------------------------------------------------------------
__global__ __launch_bounds__(256) void k_jreg(
    const float* __restrict__ Jreg, const float* __restrict__ shapedirs,
    const float* __restrict__ vtempl, float* __restrict__ JS,
    float* __restrict__ JT)
{
  const int jc = blockIdx.x;
  const int j = jc / 3, c = jc % 3;
  const int tid = threadIdx.x;
  float acc[NB_ + 1];
#pragma unroll
  for (int l = 0; l <= NB_; ++l) acc[l] = 0.f;
  for (int v = tid; v < V_; v += 256) {
    const float r = Jreg[(size_t)j * V_ + v];
    const float* sd = shapedirs + (size_t)v * 30 + c * NB_;
#pragma unroll
    for (int l = 0; l < NB_; ++l) acc[l] += r * sd[l];
    acc[NB_] += r * vtempl[v * 3 + c];
  }
  __shared__ float red[256];
#pragma unroll
  for (int t = 0; t <= NB_; ++t) {
    red[tid] = acc[t];
    __syncthreads();
    for (int s = 128; s > 0; s >>= 1) {
      if (tid < s) red[tid] += red[tid + s];
      __syncthreads();
    }
    if (tid == 0) {
      if (t < NB_) JS[j * 30 + c * NB_ + t] = red[0];
      else         JT[j * 3 + c] = red[0];
    }
    __syncthreads();
  }
}

// ---------------------------------------------------------------------------
// Kernel 2: per-batch pose pipeline (one thread per batch).
// Writes RT pre-transposed for the skinning WMMA B-operand:
//   RTt[b][e(16)][j(24)], entries e=0..11 (3x4 row-major, transl folded).
// ---------------------------------------------------------------------------
__global__ __launch_bounds__(128) void k_pose(
    const float* __restrict__ transl, const float* __restrict__ orient,
    const float* __restrict__ betas, const float* __restrict__ body_pose,
    const float* __restrict__ JS, const float* __restrict__ JT,
    float* __restrict__ PF, float* __restrict__ RTt,
    float* __restrict__ joints_out)
{
  const int b = blockIdx.x * blockDim.x + threadIdx.x;
  if (b >= B_) return;

  float bet[NB_];
#pragma unroll
  for (int l = 0; l < NB_; ++l) bet[l] = betas[(size_t)b * NB_ + l];

  float jp[J_][3];
#pragma unroll
  for (int j = 0; j < J_; ++j) {
#pragma unroll
    for (int c = 0; c < 3; ++c) {
      float s = JT[j * 3 + c];
#pragma unroll
      for (int l = 0; l < NB_; ++l) s += JS[j * 30 + c * NB_ + l] * bet[l];
      jp[j][c] = s;
    }
  }

  const float tx = transl[b * 3 + 0], ty = transl[b * 3 + 1],
              tz = transl[b * 3 + 2];
  constexpr int PAR[J_] = {-1, 0, 0, 0, 1, 2, 3, 4, 5, 6, 7, 8,
                           9, 9, 9, 12, 13, 14, 16, 17, 18, 19, 20, 21};
  float G[J_][12];

#pragma unroll
  for (int j = 0; j < J_; ++j) {
    float rx, ry, rz;
    if (j == 0) {
      rx = orient[b * 3 + 0]; ry = orient[b * 3 + 1]; rz = orient[b * 3 + 2];
    } else {
      const float* p = body_pose + (size_t)b * 69 + (j - 1) * 3;
      rx = p[0]; ry = p[1]; rz = p[2];
    }
    const float ax = rx + 1e-8f, ay = ry + 1e-8f, az = rz + 1e-8f;
    const float ang = sqrtf(ax * ax + ay * ay + az * az);
    const float inv = 1.f / ang;
    const float kx = rx * inv, ky = ry * inv, kz = rz * inv;
    const float s = sinf(ang), cc = 1.f - cosf(ang);
    float R[9];
    R[0] = 1.f - cc * (ky * ky + kz * kz);
    R[1] = -s * kz + cc * (kx * ky);
    R[2] =  s * ky + cc * (kx * kz);
    R[3] =  s * kz + cc * (kx * ky);
    R[4] = 1.f - cc * (kx * kx + kz * kz);
    R[5] = -s * kx + cc * (ky * kz);
    R[6] = -s * ky + cc * (kx * kz);
    R[7] =  s * kx + cc * (ky * kz);
    R[8] = 1.f - cc * (kx * kx + ky * ky);

    if (j > 0) {
      float* pf = PF + (size_t)b * KP + (j - 1) * 9;
      pf[0] = R[0] - 1.f; pf[1] = R[1];       pf[2] = R[2];
      pf[3] = R[3];       pf[4] = R[4] - 1.f; pf[5] = R[5];
      pf[6] = R[6];       pf[7] = R[7];       pf[8] = R[8] - 1.f;
    }

    float rl0, rl1, rl2;
    if (j == 0) { rl0 = jp[0][0]; rl1 = jp[0][1]; rl2 = jp[0][2]; }
    else {
      const int p = PAR[j];
      rl0 = jp[j][0] - jp[p][0]; rl1 = jp[j][1] - jp[p][1];
      rl2 = jp[j][2] - jp[p][2];
    }

    if (j == 0) {
      G[0][0] = R[0]; G[0][1] = R[1]; G[0][2] = R[2];  G[0][3]  = rl0;
      G[0][4] = R[3]; G[0][5] = R[4]; G[0][6] = R[5];  G[0][7]  = rl1;
      G[0][8] = R[6]; G[0][9] = R[7]; G[0][10] = R[8]; G[0][11] = rl2;
    } else {
      const int p = PAR[j];
#pragma unroll
      for (int i = 0; i < 3; ++i) {
        const float g0 = G[p][i * 4 + 0], g1 = G[p][i * 4 + 1],
                    g2 = G[p][i * 4 + 2], g3 = G[p][i * 4 + 3];
        G[j][i * 4 + 0] = g0 * R[0] + g1 * R[3] + g2 * R[6];
        G[j][i * 4 + 1] = g0 * R[1] + g1 * R[4] + g2 * R[7];
        G[j][i * 4 + 2] = g0 * R[2] + g1 * R[5] + g2 * R[8];
        G[j][i * 4 + 3] = g0 * rl0 + g1 * rl1 + g2 * rl2 + g3;
      }
    }

    joints_out[((size_t)b * J_ + j) * 3 + 0] = G[j][3]  + tx;
    joints_out[((size_t)b * J_ + j) * 3 + 1] = G[j][7]  + ty;
    joints_out[((size_t)b * J_ + j) * 3 + 2] = G[j][11] + tz;

    const float jx = jp[j][0], jy = jp[j][1], jz = jp[j][2];
    float* rt = RTt + (size_t)b * RTS + j;      // entry e lives at rt[e*24]
    rt[0 * 24]  = G[j][0]; rt[1 * 24] = G[j][1]; rt[2 * 24] = G[j][2];
    rt[3 * 24]  = G[j][3] - (G[j][0] * jx + G[j][1] * jy + G[j][2] * jz) + tx;
    rt[4 * 24]  = G[j][4]; rt[5 * 24] = G[j][5]; rt[6 * 24] = G[j][6];
    rt[7 * 24]  = G[j][7] - (G[j][4] * jx + G[j][5] * jy + G[j][6] * jz) + ty;
    rt[8 * 24]  = G[j][8]; rt[9 * 24] = G[j][9]; rt[10 * 24] = G[j][10];
    rt[11 * 24] = G[j][11] - (G[j][8] * jx + G[j][9] * jy + G[j][10] * jz) + tz;
  }
  PF[(size_t)b * KP + 207] = 0.f;
}

// ---------------------------------------------------------------------------
// Kernel 3: fused pose-GEMM + blend shapes + WMMA skinning + store.
// Block = 256 threads = 8 waves, covers 16 batches x 128 vertices.
// Tiles staged via CDNA5 async global->LDS b128 copies (ASYNCcnt).
// ---------------------------------------------------------------------------
__global__ __launch_bounds__(256) void k_verts(
    const float* __restrict__ PF, const float* __restrict__ RTt,
    const float* __restrict__ posedirs, const float* __restrict__ shapedirs,
    const float* __restrict__ vtempl, const float* __restrict__ betas,
    const float* __restrict__ lbsw, float* __restrict__ verts)
{
  __shared__ __align__(16) float rtl[16 * RTS];     // [b][e(16)][j(24)], 24 KB
  __shared__ __align__(16) float pfl[16 * KP];      // [b][208], 13 KB
  __shared__ __align__(16) float btl[16 * NB_];     // betas tile, 640 B
  __shared__ __align__(16) float tl[8 * 2 * 16 * 16];  // per-wave T tiles, 16 KB

  const int tid = threadIdx.x;
  const int lane = tid & 31;
  const int wv = tid >> 5;
  const int n = lane & 15, hi = lane >> 4;
  const int M0 = blockIdx.y * 16;
  const int N0 = (blockIdx.x * 8 + wv) * 16;

  // ---- async staging: straight 16B-granular copies into LDS ----
  {
    const unsigned rtlo = lds_off32(rtl);
    const float4* g = (const float4*)(RTt + (size_t)M0 * RTS);
    for (int i = tid; i < 16 * RTS / 4; i += 256)       // 1536 chunks
      async_b128(rtlo + 16u * i, g + i);
  }
  {
    const unsigned pflo = lds_off32(pfl);
    const float4* g = (const float4*)(PF + (size_t)M0 * KP);
    for (int i = tid; i < 16 * KP / 4; i += 256)        // 832 chunks
      async_b128(pflo + 16u * i, g + i);
  }
  {
    const unsigned btlo = lds_off32(btl);
    const float4* g = (const float4*)(betas + (size_t)M0 * NB_);
    for (int i = tid; i < 16 * NB_ / 4; i += 256)       // 40 chunks
      async_b128(btlo + 16u * i, g + i);
  }
  wait_async0();
  __syncthreads();

  const int v = N0 + n;
  const bool valid = v < V_;
  const int vcl = valid ? v : (V_ - 1);
  const size_t col = (size_t)vcl * 3;

  // ---- stage A: pose-offset GEMM, D = PF(16b x 208) x posedirs(208 x 16vc) ----
  v8f c0 = {}, c1 = {}, c2 = {};
  const float* pfr = pfl + n * KP;
  for (int kk = 0; kk < 52; ++kk) {
    const int kb = kk * 4 + hi * 2;
    const int kb1 = (kb + 1 <= 206) ? (kb + 1) : 206;  // A col 207 is 0-pad
    v2f a; a.x = pfr[kb]; a.y = pfr[kb + 1];
    const float* p0 = posedirs + (size_t)kb * VC + col;
    const float* p1 = posedirs + (size_t)kb1 * VC + col;
    v2f b0, b1, b2;
    b0.x = p0[0]; b1.x = p0[1]; b2.x = p0[2];
    b0.y = p1[0]; b1.y = p1[1]; b2.y = p1[2];
    c0 = __builtin_amdgcn_wmma_f32_16x16x4_f32(false, a, false, b0, (short)0,
                                               c0, false, false);
    c1 = __builtin_amdgcn_wmma_f32_16x16x4_f32(false, a, false, b1, (short)0,
                                               c1, false, false);
    c2 = __builtin_amdgcn_wmma_f32_16x16x4_f32(false, a, false, b2, (short)0,
                                               c2, false, false);
  }

  // ---- v_posed per (batch row, lane) ----
  float sd[30];
#pragma unroll
  for (int i = 0; i < 30; ++i) sd[i] = shapedirs[(size_t)vcl * 30 + i];
  const float vt0 = vtempl[vcl * 3 + 0];
  const float vt1 = vtempl[vcl * 3 + 1];
  const float vt2 = vtempl[vcl * 3 + 2];

  float pvx[8], pvy[8], pvz[8];
#pragma unroll
  for (int r = 0; r < 8; ++r) {
    const int bl = r + 8 * hi;
    const float* bet = btl + bl * NB_;
    float x = vt0 + c0[r], y = vt1 + c1[r], z = vt2 + c2[r];
#pragma unroll
    for (int l = 0; l < NB_; ++l) {
      const float be = bet[l];
      x += sd[l] * be; y += sd[10 + l] * be; z += sd[20 + l] * be;
    }
    pvx[r] = x; pvy[r] = y; pvz[r] = z;
  }

  // ---- stage B: WMMA skinning  T = W(16v x 24j) x RTt_b(24j x 16e) ----
  float w[J_];
#pragma unroll
  for (int j = 0; j < J_; ++j) w[j] = lbsw[(size_t)vcl * J_ + j];

  v2f aw[6];                      // loop-invariant A-operand pairs
#pragma unroll
  for (int s = 0; s < 6; ++s) {
    aw[s].x = hi ? w[4 * s + 2] : w[4 * s + 0];
    aw[s].y = hi ? w[4 * s + 3] : w[4 * s + 1];
  }

  float* tb0 = tl + (wv * 2 + 0) * 256;
  float* tb1 = tl + (wv * 2 + 1) * 256;

#pragma unroll
  for (int r2 = 0; r2 < 8; ++r2) {
    v8f t0 = {}, t1 = {};
#pragma unroll
    for (int s = 0; s < 6; ++s) {
      const int j0 = 4 * s + 2 * hi;
      const v2f bb0 = *(const v2f*)&rtl[(r2 * 16 + n) * 24 + j0];
      const v2f bb1 = *(const v2f*)&rtl[((r2 + 8) * 16 + n) * 24 + j0];
      t0 = __builtin_amdgcn_wmma_f32_16x16x4_f32(false, aw[s], false, bb0,
                                                 (short)0, t0, false, false);
      t1 = __builtin_amdgcn_wmma_f32_16x16x4_f32(false, aw[s], false, bb1,
                                                 (short)0, t1, false, false);
    }
    // bounce T tiles through LDS: row m = r+8*hi, col e = n
#pragma unroll
    for (int r = 0; r < 8; ++r) {
      tb0[(r + 8 * hi) * 16 + n] = t0[r];
      tb1[(r + 8 * hi) * 16 + n] = t1[r];
    }
    // apply: lane (hi,n) handles batch M0 + r2 + 8*hi, vertex v
    // (same-wave LDS ops execute in order -> RAW through LDS is safe)
    const float* tp = (hi ? tb1 : tb0) + n * 16;
    const float4 q0 = *(const float4*)(tp + 0);
    const float4 q1 = *(const float4*)(tp + 4);
    const float4 q2 = *(const float4*)(tp + 8);
    if (valid) {
      float* o = verts + ((size_t)(M0 + r2 + 8 * hi) * V_ + v) * 3;
      o[0] = q0.x * pvx[r2] + q0.y * pvy[r2] + q0.z * pvz[r2] + q0.w;
      o[1] = q1.x * pvx[r2] + q1.y * pvy[r2] + q1.z * pvz[r2] + q1.w;
      o[2] = q2.x * pvx[r2] + q2.y * pvy[r2] + q2.z * pvz[r2] + q2.w;
    }
  }
}

// ---------------------------------------------------------------------------
// Workspace layout (floats):
//   JS[720] | JT[72] | pad | PF[1024*208] | RTt[1024*384]   (~2.4 MB)
// ---------------------------------------------------------------------------
static constexpr size_t JS_OFF = 0;
static constexpr size_t JT_OFF = 720;
static constexpr size_t PF_OFF = 1024;
static constexpr size_t RT_OFF = PF_OFF + (size_t)B_ * KP;

extern "C" void kernel_launch(void* const* d_in, const int* in_sizes, int n_in,
                              void* d_out, int out_size, void* d_ws,
                              size_t ws_size, hipStream_t stream) {
  const float* transl    = (const float*)d_in[0];
  const float* orient    = (const float*)d_in[1];
  const float* betas     = (const float*)d_in[2];
  const float* body_pose = (const float*)d_in[3];
  const float* vtempl    = (const float*)d_in[4];
  const float* shapedirs = (const float*)d_in[5];
  const float* posedirs  = (const float*)d_in[6];
  const float* Jreg      = (const float*)d_in[7];
  const float* lbsw      = (const float*)d_in[8];
  // d_in[9] = parents: fixed SMPL topology, hardcoded in k_pose.

  float* ws = (float*)d_ws;
  float* verts = (float*)d_out;
  float* joints = verts + (size_t)B_ * V_ * 3;

  k_jreg<<<72, 256, 0, stream>>>(Jreg, shapedirs, vtempl, ws + JS_OFF,
                                 ws + JT_OFF);
  k_pose<<<8, 128, 0, stream>>>(transl, orient, betas, body_pose, ws + JS_OFF,
                                ws + JT_OFF, ws + PF_OFF, ws + RT_OFF, joints);
  dim3 g3(54, 64);  // 54*8 = 432 vertex tiles (431 needed), 64 batch tiles
  k_verts<<<g3, 256, 0, stream>>>(ws + PF_OFF, ws + RT_OFF, posedirs,
                                  shapedirs, vtempl, betas, lbsw, verts);
}